// HDRnetModel_90177133347159
// MI455X (gfx1250) — compile-verified
//
#include <hip/hip_runtime.h>
#include <hip/hip_bf16.h>
#include <stdint.h>

typedef __attribute__((ext_vector_type(2))) float v2f;
typedef __attribute__((ext_vector_type(8))) float v8f;

#define LEAKF 0.01f
#define IMH 1024
#define IMW 1024
#define CG 12          // grid channels (3x4 affine)
#define DG 12          // grid depth
#define GGH 16
#define GGW 16
#define TILE 32
#define FT_ROW 36            // padded row stride for fullres tile (34 + 2 pad)
#define FT_CH (34 * FT_ROW)  // per-channel stride

__device__ __forceinline__ int imin(int a, int b) { return a < b ? a : b; }

__global__ __launch_bounds__(256)
void hdrnet_fused_kernel(const float* __restrict__ grid,
                         const float* __restrict__ fullres,
                         const float* __restrict__ w1,
                         const float* __restrict__ w3,
                         const float* __restrict__ b3,
                         const float* __restrict__ gbias,
                         float* __restrict__ out)
{
    // LDS: fullres tile w/ halo (3*34 rows, stride 36) + reordered grid window + guide tile
    __shared__ __align__(16) float ftile[3 * FT_CH];       // 3672 f32
    __shared__ __align__(16) float gwin[DG * 3 * 3 * CG];  // 1296 f32, [z][yy][xx][c]
    __shared__ float gtile[TILE * TILE];                   // 1024 f32

    const int tid = threadIdx.x;
    const int n   = blockIdx.z;
    const int tx0 = blockIdx.x * TILE;
    const int ty0 = blockIdx.y * TILE;
    const float S = 15.0f / 1023.0f;   // pixel -> grid coordinate scale
    const int gy0 = (int)(ty0 * S);    // same float expr as per-pixel => containment
    const int gx0 = (int)(tx0 * S);

    // ---- async stage: fullres tile + halo (zero pad outside image) ----
    const float* fr = fullres + (size_t)n * 3 * IMH * IMW;
    for (int e = tid; e < 3 * 34 * 34; e += 256) {
        int ch = e / (34 * 34);
        int r  = e - ch * (34 * 34);
        int yy = r / 34;
        int xx = r - yy * 34;
        int gy = ty0 - 1 + yy;
        int gx = tx0 - 1 + xx;
        float* dstp = &ftile[ch * FT_CH + yy * FT_ROW + xx];
        if ((unsigned)gy < (unsigned)IMH && (unsigned)gx < (unsigned)IMW) {
            const float* srcp = fr + ((size_t)ch * IMH + gy) * IMW + gx;
            unsigned doff = (unsigned)(uintptr_t)dstp;
            asm volatile("global_load_async_to_lds_b32 %0, %1, off"
                         :: "v"(doff), "v"((unsigned long long)(uintptr_t)srcp)
                         : "memory");
        } else {
            *dstp = 0.0f;  // SAME conv zero padding
        }
    }

    // ---- async stage: 3x3 grid window over all (c,z), reordered to [z][yy][xx][c] ----
    const float* gr = grid + (size_t)n * CG * DG * GGH * GGW;
    for (int e = tid; e < DG * 9 * CG; e += 256) {
        int c = e % CG;
        int q = e / CG;
        int xx = q % 3; q /= 3;
        int yy = q % 3;
        int z  = q / 3;
        int gy = imin(gy0 + yy, GGH - 1);
        int gx = imin(gx0 + xx, GGW - 1);
        const float* srcp = gr + (((size_t)c * DG + z) * GGH + gy) * GGW + gx;
        unsigned doff = (unsigned)(uintptr_t)&gwin[e];
        asm volatile("global_load_async_to_lds_b32 %0, %1, off"
                     :: "v"(doff), "v"((unsigned long long)(uintptr_t)srcp)
                     : "memory");
    }

    asm volatile("s_wait_asynccnt 0" ::: "memory");
    __syncthreads();

    // ---- per-lane constant WMMA operands ----
    // A (16x28, K padded 27->28): rows 0..2 = w3 im2col, rows 3..5 = w1 at center tap.
    // Layout (f32 WMMA 16x16x4): lane L holds M=L&15; K = 4*s + 2*(L>>4) + {0,1}.
    const int lane = tid & 31;
    const int m   = lane & 15;
    const int hh2 = lane >> 4;

    v2f aop[7];
    int boff[7][2];
#pragma unroll
    for (int s = 0; s < 7; ++s) {
#pragma unroll
        for (int e = 0; e < 2; ++e) {
            int kk = 4 * s + 2 * hh2 + e;
            float av = 0.0f;
            int off = 0;                  // padded K: A is zero there, B value irrelevant
            if (kk < 27) {
                int ch = kk / 9;
                int rr = kk - ch * 9;
                int dy = rr / 3;
                int dx = rr - dy * 3;
                off = ch * FT_CH + dy * FT_ROW + dx;
                if (m < 3)                 av = w3[m * 27 + kk];        // OIHW flat
                else if (m < 6 && rr == 4) av = w1[(m - 3) * 3 + ch];   // center tap
            }
            aop[s][e]  = av;
            boff[s][e] = off;
        }
    }

    const float b30 = b3[0], b31 = b3[1], b32 = b3[2];
    const float gb  = gbias[0];

    // ---- guide via WMMA: 8 waves x 8 groups of 16 pixels = 1024 pixels ----
    const int wave = tid >> 5;
#pragma unroll
    for (int gi = 0; gi < 8; ++gi) {
        int g  = wave * 8 + gi;
        int py = g >> 1;
        int pxg = ((g & 1) << 4) + m;            // B column N = lane&15
        int pbase = py * FT_ROW + pxg;           // patch origin in halo coords
        v8f acc = {0.f, 0.f, 0.f, 0.f, 0.f, 0.f, 0.f, 0.f};
#pragma unroll
        for (int s = 0; s < 7; ++s) {
            v2f bop;
            bop[0] = ftile[pbase + boff[s][0]];
            bop[1] = ftile[pbase + boff[s][1]];
            acc = __builtin_amdgcn_wmma_f32_16x16x4_f32(false, aop[s], false, bop,
                                                        (short)0, acc, false, false);
        }
        // lanes 0..15 hold rows 0..7 of their column => all 6 conv outputs
        if (hh2 == 0) {
            float t, ssum = 0.0f;
            t = acc[0] + b30; ssum += (t > 0.0f) ? t : t * LEAKF;
            t = acc[1] + b31; ssum += (t > 0.0f) ? t : t * LEAKF;
            t = acc[2] + b32; ssum += (t > 0.0f) ? t : t * LEAKF;
            t = acc[3];       ssum += (t > 0.0f) ? t : t * LEAKF;
            t = acc[4];       ssum += (t > 0.0f) ? t : t * LEAKF;
            t = acc[5];       ssum += (t > 0.0f) ? t : t * LEAKF;
            float gv = ssum * (1.0f / 3.0f) + gb;
            gv = fminf(fmaxf(gv, 0.0f), 1.0f);
            gtile[py * TILE + pxg] = gv;
        }
    }
    __syncthreads();

    // ---- slice + apply: 4 pixels per thread, coalesced stores ----
    const int px2 = tid & 31;
    const int pyb = tid >> 5;
#pragma unroll
    for (int it = 0; it < 4; ++it) {
        int py = pyb + it * 8;
        int y = ty0 + py;
        int x = tx0 + px2;
        float fy = y * S; int iy0 = (int)fy; float tyf = fy - (float)iy0;
        float fx = x * S; int ix0 = (int)fx; float txf = fx - (float)ix0;
        float gv = gtile[py * TILE + px2];
        float fz = gv * 11.0f; int iz0 = (int)fz; float tzf = fz - (float)iz0;

        int zs[2] = { iz0, imin(iz0 + 1, DG - 1) };
        int ys[2] = { iy0 - gy0, imin(iy0 + 1, GGH - 1) - gy0 };
        int xs[2] = { ix0 - gx0, imin(ix0 + 1, GGW - 1) - gx0 };
        float wzv[2] = { 1.0f - tzf, tzf };
        float wyv[2] = { 1.0f - tyf, tyf };
        float wxv[2] = { 1.0f - txf, txf };

        float cf[12];
#pragma unroll
        for (int j = 0; j < 12; ++j) cf[j] = 0.0f;

#pragma unroll
        for (int dz = 0; dz < 2; ++dz)
#pragma unroll
        for (int dy = 0; dy < 2; ++dy)
#pragma unroll
        for (int dx = 0; dx < 2; ++dx) {
            float wgt = wzv[dz] * wyv[dy] * wxv[dx];
            const float4* p = (const float4*)&gwin[((zs[dz] * 3 + ys[dy]) * 3 + xs[dx]) * CG];
            float4 v0 = p[0], v1 = p[1], v2 = p[2];
            cf[0] += wgt * v0.x; cf[1] += wgt * v0.y; cf[2]  += wgt * v0.z; cf[3]  += wgt * v0.w;
            cf[4] += wgt * v1.x; cf[5] += wgt * v1.y; cf[6]  += wgt * v1.z; cf[7]  += wgt * v1.w;
            cf[8] += wgt * v2.x; cf[9] += wgt * v2.y; cf[10] += wgt * v2.z; cf[11] += wgt * v2.w;
        }

        int cidx = (py + 1) * FT_ROW + (px2 + 1);  // center pixel in halo coords
        float R = ftile[0 * FT_CH + cidx];
        float G = ftile[1 * FT_CH + cidx];
        float B = ftile[2 * FT_CH + cidx];
        float o0 = cf[0] * R + cf[1] * G + cf[2]  * B + cf[3];
        float o1 = cf[4] * R + cf[5] * G + cf[6]  * B + cf[7];
        float o2 = cf[8] * R + cf[9] * G + cf[10] * B + cf[11];

        out[(((size_t)n * 3 + 0) * IMH + y) * IMW + x] = o0;
        out[(((size_t)n * 3 + 1) * IMH + y) * IMW + x] = o1;
        out[(((size_t)n * 3 + 2) * IMH + y) * IMW + x] = o2;
    }
}

extern "C" void kernel_launch(void* const* d_in, const int* in_sizes, int n_in,
                              void* d_out, int out_size, void* d_ws, size_t ws_size,
                              hipStream_t stream) {
    const float* grid    = (const float*)d_in[0];
    const float* fullres = (const float*)d_in[1];
    const float* w1      = (const float*)d_in[2];
    const float* w3      = (const float*)d_in[3];
    const float* b3      = (const float*)d_in[4];
    const float* gbias   = (const float*)d_in[5];
    float* outp = (float*)d_out;
    dim3 blocks(IMW / TILE, IMH / TILE, 4);
    hdrnet_fused_kernel<<<blocks, 256, 0, stream>>>(grid, fullres, w1, w3, b3, gbias, outp);
}